// UpSample_5446018532081
// MI455X (gfx1250) — compile-verified
//
#include <hip/hip_runtime.h>
#include <hip/hip_bf16.h>
#include <cstddef>

// ---------------- problem constants ----------------
#define BATCH   2
#define F_IN    256
#define L_IN    250
#define STRIDE  8
#define KSIZE   16
#define F_OUT   128
#define L_SEQ   2008            // (L_IN-1)*STRIDE + KSIZE
#define ROWS    (BATCH * L_SEQ) // 4016 (multiple of 16)
#define D_INNER 256
#define D_STATE 16
#define DT_RANK 8
#define XDBL_W  (DT_RANK + 2 * D_STATE) // 40
#define LN_EPS  1e-5f
#define GEMM_NT 4               // N-subtiles per wave (16x64 C strip)

typedef __attribute__((ext_vector_type(2))) float v2f;
typedef __attribute__((ext_vector_type(8))) float v8f;

__device__ __forceinline__ float siluf(float x) {
    return x / (1.f + __expf(-x));
}
__device__ __forceinline__ float softplusf(float x) {
    return (x > 20.f) ? x : logf(1.f + __expf(x));
}

// ---------------------------------------------------------------------------
// Generic fp32 GEMM on the CDNA5 WMMA f32 path:
//   C[M,N] = act(A[M,K] * W[N,K]^T + bias[N])
// One wave32 per 16x(16*GEMM_NT) C strip. K consumed 4 at a time with
// V_WMMA_F32_16X16X4_F32; the A fragment is loaded once per K-step and
// reused across GEMM_NT WMMAs (register-level N blocking).
// Fragments (per CDNA5 ISA):
//   A 16x4 f32 : lane(m + 16h) holds k = 2h, 2h+1 in vgpr0/1 -> one float2
//   B 4x16 f32 : mirror with n in place of m
//   C/D 16x16  : vgpr v -> M = v + 8h, N = lane & 15
// All branches before/around WMMA are wave-uniform; partial-N subtiles clamp
// load rows and predicate only the stores, so EXEC is all-1s at every WMMA.
// ---------------------------------------------------------------------------
template <int ACT> // 0 none, 1 silu, 2 softplus
__global__ void gemm_wmma_f32(const float* __restrict__ A, int lda,
                              const float* __restrict__ W, int ldw,
                              const float* __restrict__ bias,
                              float* __restrict__ C, int ldc,
                              int M, int N, int K) {
    const int wave = threadIdx.x >> 5;
    const int lane = threadIdx.x & 31;
    const int tiles_n  = (N + 15) >> 4;
    const int tiles_m  = (M + 15) >> 4;
    const int groups_n = (tiles_n + GEMM_NT - 1) / GEMM_NT;
    const int task = blockIdx.x * (blockDim.x >> 5) + wave;
    if (task >= tiles_m * groups_n) return;   // wave-uniform
    const int tm = task / groups_n;
    const int tg = task - tm * groups_n;

    const int h  = lane >> 4;   // k-half selector
    const int ml = lane & 15;

    const int am = tm * 16 + ml;              // A row (M multiple of 16 here)
    const float* Ap = A + (size_t)am * lda + 2 * h;

    const float* Wp[GEMM_NT];
    bool  nvalid[GEMM_NT];
    int   wn[GEMM_NT];
#pragma unroll
    for (int j = 0; j < GEMM_NT; ++j) {
        wn[j] = (tg * GEMM_NT + j) * 16 + ml; // W row == output column
        nvalid[j] = (wn[j] < N);
        const int wn_c = nvalid[j] ? wn[j] : 0; // clamp loads, predicate stores
        Wp[j] = W + (size_t)wn_c * ldw + 2 * h;
    }

    v8f acc[GEMM_NT];
#pragma unroll
    for (int j = 0; j < GEMM_NT; ++j) acc[j] = (v8f){};

    for (int k = 0; k < K; k += 4) {
        const v2f a = *(const v2f*)(Ap + k);   // shared across all N subtiles
#pragma unroll
        for (int j = 0; j < GEMM_NT; ++j) {
            const v2f b = *(const v2f*)(Wp[j] + k);
            acc[j] = __builtin_amdgcn_wmma_f32_16x16x4_f32(
                /*neg_a=*/false, a, /*neg_b=*/false, b,
                /*c_mod=*/(short)0, acc[j], /*reuse_a=*/false, /*reuse_b=*/false);
        }
    }

#pragma unroll
    for (int j = 0; j < GEMM_NT; ++j) {
        if (!nvalid[j]) continue;
        const float bval = bias ? bias[wn[j]] : 0.f;
#pragma unroll
        for (int v = 0; v < 8; ++v) {
            const int m = tm * 16 + v + 8 * h;
            float xv = acc[j][v] + bval;
            if (ACT == 1) xv = siluf(xv);
            if (ACT == 2) xv = softplusf(xv);
            C[(size_t)m * ldc + wn[j]] = xv;
        }
    }
}

// ---------------------------------------------------------------------------
// ConvTranspose1d(256->128, k=16, s=8) + bias + SiLU, fused with channel
// concat:  cat[b*L+l][0:128] = skip, cat[..][128:256] = silu(convT).
// One block per (b,l); the two contributing x columns are staged in LDS.
// ---------------------------------------------------------------------------
__global__ void convt_cat_kernel(const float* __restrict__ x,
                                 const float* __restrict__ w,   // (256,128,16)
                                 const float* __restrict__ cb,  // (128)
                                 const float* __restrict__ skip,// (B,128,L)
                                 float* __restrict__ cat) {
    __shared__ float x1[F_IN];
    __shared__ float x2[F_IN];
    const int bl = blockIdx.x;        // 0..ROWS-1
    const int b = bl / L_SEQ;
    const int l = bl - b * L_SEQ;
    const int t = threadIdx.x;        // 0..127
    const int k1 = l & 7;
    const int i1 = l >> 3;            // tap at kernel index k1
    const int i2 = i1 - 1;            // tap at kernel index k1+8
    for (int c = t; c < F_IN; c += F_OUT) {
        x1[c] = (i1 < L_IN) ? x[((size_t)b * F_IN + c) * L_IN + i1] : 0.f;
        x2[c] = (i2 >= 0)   ? x[((size_t)b * F_IN + c) * L_IN + i2] : 0.f;
    }
    __syncthreads();
    float s = cb[t];
    for (int c = 0; c < F_IN; ++c) {
        const float* wc = w + ((size_t)c * F_OUT + t) * KSIZE;
        s += wc[k1] * x1[c] + wc[k1 + 8] * x2[c];
    }
    cat[(size_t)bl * (2 * F_OUT) + t]         = skip[((size_t)b * F_OUT + t) * L_SEQ + l];
    cat[(size_t)bl * (2 * F_OUT) + F_OUT + t] = siluf(s);
}

// ---------------------------------------------------------------------------
// Row LayerNorm over 128 features. One 128-thread block per row.
// ---------------------------------------------------------------------------
__global__ void layernorm_kernel(const float* __restrict__ in,
                                 const float* __restrict__ w,
                                 const float* __restrict__ b,
                                 float* __restrict__ out) {
    __shared__ float sd[F_OUT];
    const int r = blockIdx.x;
    const int t = threadIdx.x;
    const float v = in[(size_t)r * F_OUT + t];
    sd[t] = v;
    __syncthreads();
    for (int s = 64; s > 0; s >>= 1) { if (t < s) sd[t] += sd[t + s]; __syncthreads(); }
    const float mu = sd[0] * (1.f / F_OUT);
    __syncthreads();
    const float dv = v - mu;
    sd[t] = dv * dv;
    __syncthreads();
    for (int s = 64; s > 0; s >>= 1) { if (t < s) sd[t] += sd[t + s]; __syncthreads(); }
    const float var = sd[0] * (1.f / F_OUT);
    out[(size_t)r * F_OUT + t] = dv * rsqrtf(var + LN_EPS) * w[t] + b[t];
}

// ---------------------------------------------------------------------------
// Depthwise causal conv (k=4) on xc = xz[:, :256] + bias + SiLU.
// ---------------------------------------------------------------------------
__global__ void dwconv_kernel(const float* __restrict__ xz,  // (ROWS,512)
                              const float* __restrict__ cw,  // (256,4)
                              const float* __restrict__ cb,  // (256)
                              float* __restrict__ u) {       // (ROWS,256)
    const int idx = blockIdx.x * blockDim.x + threadIdx.x;
    if (idx >= ROWS * D_INNER) return;
    const int r = idx >> 8;          // row
    const int d = idx & 255;
    const int b = r / L_SEQ;
    const int l = r - b * L_SEQ;
    float s = cb[d];
#pragma unroll
    for (int j = 0; j < 4; ++j) {
        const int ll = l - 3 + j;
        if (ll >= 0)
            s += cw[d * 4 + j] * xz[((size_t)b * L_SEQ + ll) * (2 * D_INNER) + d];
    }
    u[(size_t)r * D_INNER + d] = siluf(s);
}

// ---------------------------------------------------------------------------
// Selective scan. One wave32 handles two (b,d) channels (16 states each).
// Sequential over L; per step the n=16 state partials reduce with shfl_xor
// (stays inside the 16-lane half of the wave). Lane n==0 applies D-skip and
// the SiLU(z) gate and stores y.
// ---------------------------------------------------------------------------
__global__ void scan_kernel(const float* __restrict__ delta, // (ROWS,256)
                            const float* __restrict__ u,     // (ROWS,256)
                            const float* __restrict__ xdbl,  // (ROWS,40)
                            const float* __restrict__ A_log, // (256,16)
                            const float* __restrict__ Dsk,   // (256)
                            const float* __restrict__ xz,    // (ROWS,512), z at +256
                            float* __restrict__ y) {         // (ROWS,256)
    const int wave = (blockIdx.x * blockDim.x + threadIdx.x) >> 5; // 0..255
    const int lane = threadIdx.x & 31;
    const int n    = lane & 15;
    const int half = lane >> 4;
    const int ch = wave * 2 + half;   // 0..511 = b*256 + d
    const int b = ch >> 8;
    const int d = ch & 255;
    const float Av = -__expf(A_log[d * D_STATE + n]);
    const float Dv = Dsk[d];
    float h = 0.f;
    for (int l = 0; l < L_SEQ; ++l) {
        const size_t r = (size_t)b * L_SEQ + l;
        const float dt = delta[r * D_INNER + d];
        const float uv = u[r * D_INNER + d];
        const float Bv = xdbl[r * XDBL_W + DT_RANK + n];
        const float Cv = xdbl[r * XDBL_W + DT_RANK + D_STATE + n];
        h = __expf(dt * Av) * h + (dt * uv) * Bv;
        float yp = h * Cv;
        yp += __shfl_xor(yp, 8, 32);
        yp += __shfl_xor(yp, 4, 32);
        yp += __shfl_xor(yp, 2, 32);
        yp += __shfl_xor(yp, 1, 32);
        if (n == 0) {
            const float zv = xz[r * (2 * D_INNER) + D_INNER + d];
            y[r * D_INNER + d] = (yp + Dv * uv) * siluf(zv);
        }
    }
}

// ---------------------------------------------------------------------------
// Small elementwise helpers.
// ---------------------------------------------------------------------------
__global__ void add_kernel(const float* __restrict__ a, const float* __restrict__ b,
                           float* __restrict__ o, int n) {
    const int i = blockIdx.x * blockDim.x + threadIdx.x;
    if (i < n) o[i] = a[i] + b[i];
}
__global__ void copy_kernel(const float* __restrict__ a, float* __restrict__ o, int n) {
    const int i = blockIdx.x * blockDim.x + threadIdx.x;
    if (i < n) o[i] = a[i];
}
__global__ void flip_kernel(const float* __restrict__ a, float* __restrict__ o) {
    const int i = blockIdx.x * blockDim.x + threadIdx.x;
    if (i >= ROWS * F_OUT) return;
    const int r = i >> 7, c = i & 127;
    const int b = r / L_SEQ, l = r - b * L_SEQ;
    o[i] = a[((size_t)b * L_SEQ + (L_SEQ - 1 - l)) * F_OUT + c];
}
__global__ void addflip_kernel(const float* __restrict__ a, float* __restrict__ o) {
    const int i = blockIdx.x * blockDim.x + threadIdx.x;
    if (i >= ROWS * F_OUT) return;
    const int r = i >> 7, c = i & 127;
    const int b = r / L_SEQ, l = r - b * L_SEQ;
    o[i] += a[((size_t)b * L_SEQ + (L_SEQ - 1 - l)) * F_OUT + c];
}
__global__ void transpose_out_kernel(const float* __restrict__ acc, float* __restrict__ out) {
    const int i = blockIdx.x * blockDim.x + threadIdx.x;
    if (i >= ROWS * F_OUT) return;
    const int r = i >> 7, c = i & 127;
    const int b = r / L_SEQ, l = r - b * L_SEQ;
    out[((size_t)b * F_OUT + c) * L_SEQ + l] = acc[i];
}

// ---------------------------------------------------------------------------
extern "C" void kernel_launch(void* const* d_in, const int* in_sizes, int n_in,
                              void* d_out, int out_size, void* d_ws, size_t ws_size,
                              hipStream_t stream) {
    (void)in_sizes; (void)n_in; (void)out_size; (void)ws_size;
    const float* x        = (const float*)d_in[0];
    const float* skip     = (const float*)d_in[1];
    const float* convT_w  = (const float*)d_in[4];
    const float* convT_b  = (const float*)d_in[5];
    const float* fuse_w   = (const float*)d_in[6];
    const float* fuse_b   = (const float*)d_in[7];
    const float* norm_w   = (const float*)d_in[8];
    const float* norm_b   = (const float*)d_in[9];
    const float* in_proj  = (const float*)d_in[10];
    const float* conv_w   = (const float*)d_in[11];
    const float* conv_b   = (const float*)d_in[12];
    const float* x_proj   = (const float*)d_in[13];
    const float* dt_w     = (const float*)d_in[14];
    const float* dt_b     = (const float*)d_in[15];
    const float* A_log    = (const float*)d_in[16];
    const float* D_skip   = (const float*)d_in[17];
    const float* out_proj = (const float*)d_in[18];
    float* out = (float*)d_out;

    // workspace layout (floats)
    float* ws = (float*)d_ws;
    float* CAT   = ws;                      // ROWS*256
    float* XS    = CAT   + (size_t)ROWS * 256;
    float* XSB   = XS    + (size_t)ROWS * F_OUT;
    float* LN    = XSB   + (size_t)ROWS * F_OUT;
    float* XZ    = LN    + (size_t)ROWS * F_OUT;   // ROWS*512
    float* U     = XZ    + (size_t)ROWS * 512;     // ROWS*256
    float* XDBL  = U     + (size_t)ROWS * 256;     // ROWS*40
    float* DELTA = XDBL  + (size_t)ROWS * XDBL_W;  // ROWS*256
    float* Y     = DELTA + (size_t)ROWS * 256;     // ROWS*256
    float* M0    = Y     + (size_t)ROWS * 256;     // ROWS*128
    float* R2    = M0    + (size_t)ROWS * F_OUT;
    float* M1    = R2    + (size_t)ROWS * F_OUT;
    float* ACC   = M1    + (size_t)ROWS * F_OUT;

    const int NEL  = ROWS * F_OUT;
    const int EB   = (NEL + 255) / 256;

    auto gemm = [&](int act, const float* A, int lda, const float* W, int ldw,
                    const float* bias, float* C, int ldc, int M, int N, int K) {
        const int tiles_m  = (M + 15) / 16;
        const int groups_n = (((N + 15) / 16) + GEMM_NT - 1) / GEMM_NT;
        const int tasks  = tiles_m * groups_n;
        const int blocks = (tasks + 7) / 8;     // 8 waves / 256-thread block
        if (act == 0)
            gemm_wmma_f32<0><<<blocks, 256, 0, stream>>>(A, lda, W, ldw, bias, C, ldc, M, N, K);
        else if (act == 1)
            gemm_wmma_f32<1><<<blocks, 256, 0, stream>>>(A, lda, W, ldw, bias, C, ldc, M, N, K);
        else
            gemm_wmma_f32<2><<<blocks, 256, 0, stream>>>(A, lda, W, ldw, bias, C, ldc, M, N, K);
    };

    // Front-end: convT + SiLU + concat, then fuse GEMM (SiLU) -> xs (ROWS,128)
    convt_cat_kernel<<<ROWS, F_OUT, 0, stream>>>(x, convT_w, convT_b, skip, CAT);
    gemm(1, CAT, 256, fuse_w, 256, fuse_b, XS, F_OUT, ROWS, F_OUT, 256);
    flip_kernel<<<EB, 256, 0, stream>>>(XS, XSB);

    auto mixer = [&](int dir, int blk, const float* lnin, float* mout) {
        const int di = dir * 2 + blk;
        gemm(0, lnin, F_OUT, in_proj + (size_t)di * 512 * F_OUT, F_OUT,
             nullptr, XZ, 512, ROWS, 512, F_OUT);
        dwconv_kernel<<<(ROWS * D_INNER + 255) / 256, 256, 0, stream>>>(
            XZ, conv_w + (size_t)di * D_INNER * 4, conv_b + (size_t)di * D_INNER, U);
        gemm(0, U, 256, x_proj + (size_t)di * XDBL_W * 256, 256,
             nullptr, XDBL, XDBL_W, ROWS, XDBL_W, 256);
        gemm(2, XDBL, XDBL_W, dt_w + (size_t)di * 256 * DT_RANK, DT_RANK,
             dt_b + (size_t)di * 256, DELTA, 256, ROWS, 256, DT_RANK);
        scan_kernel<<<32, 256, 0, stream>>>(DELTA, U, XDBL,
            A_log + (size_t)di * 256 * D_STATE, D_skip + (size_t)di * 256, XZ, Y);
        gemm(0, Y, 256, out_proj + (size_t)di * F_OUT * 256, 256,
             nullptr, mout, F_OUT, ROWS, F_OUT, 256);
    };

    for (int dir = 0; dir < 2; ++dir) {
        const float* xsd = dir ? XSB : XS;
        // block 0
        layernorm_kernel<<<ROWS, F_OUT, 0, stream>>>(
            xsd, norm_w + (size_t)(dir * 2 + 0) * F_OUT,
            norm_b + (size_t)(dir * 2 + 0) * F_OUT, LN);
        mixer(dir, 0, LN, M0);
        add_kernel<<<EB, 256, 0, stream>>>(M0, xsd, R2, NEL);
        // block 1
        layernorm_kernel<<<ROWS, F_OUT, 0, stream>>>(
            R2, norm_w + (size_t)(dir * 2 + 1) * F_OUT,
            norm_b + (size_t)(dir * 2 + 1) * F_OUT, LN);
        mixer(dir, 1, LN, M1);
        add_kernel<<<EB, 256, 0, stream>>>(M1, R2, M0, NEL); // M0 = stack output
        if (dir == 0)
            copy_kernel<<<EB, 256, 0, stream>>>(M0, ACC, NEL);
        else
            addflip_kernel<<<EB, 256, 0, stream>>>(M0, ACC);
    }

    transpose_out_kernel<<<EB, 256, 0, stream>>>(ACC, out);
}